// MultiGraphAttention_32023276159483
// MI455X (gfx1250) — compile-verified
//
#include <hip/hip_runtime.h>
#include <cfloat>

// Problem constants (match reference)
#define B_   64
#define N_   1024
#define D_   3
#define H_   4
#define F_   16
#define K_   20
#define BN_EPS 1e-3f

typedef __attribute__((ext_vector_type(2))) float v2f;
typedef __attribute__((ext_vector_type(8))) float v8f;

// ---------------------------------------------------------------------------
// Kernel 1: fused pairwise-distance (WMMA f32 16x16x4 Gram tiles) + top-20.
// Grid: (N/16 row tiles, B batches), block = 128 threads = 4 waves.
// Each wave computes adj for the block's 16 rows x its 256-column slice via
// 16 WMMA tiles, writes into a 16x1024 LDS strip; then each wave selects the
// 20 smallest entries for 4 rows with shuffle-reduced argmin.
// ---------------------------------------------------------------------------
__global__ __launch_bounds__(128) void knn_topk_kernel(
    const float* __restrict__ pc, int* __restrict__ knnIdx)
{
    __shared__ float adjS[16 * N_];       // 64 KB (WGP has 320 KB)

    const int wave = threadIdx.x >> 5;
    const int lane = threadIdx.x & 31;
    const int hi   = lane >> 4;           // lane half (WMMA K split)
    const int lm   = lane & 15;
    const int row0 = blockIdx.x * 16;
    const int b    = blockIdx.y;
    const float* pcb = pc + (size_t)b * N_ * 3;

    // A operand: rows row0..row0+15, K = coord dim padded to 4.
    // 32-bit A 16x4 layout: lanes 0-15 hold K=0,1 ; lanes 16-31 hold K=2,3.
    const int ar = row0 + lm;
    const float rx = pcb[ar*3+0], ry = pcb[ar*3+1], rz = pcb[ar*3+2];
    v2f a;
    a.x = hi ? rz : rx;
    a.y = hi ? 0.0f : ry;
    const float sqLane = rx*rx + ry*ry + rz*rz;   // |row (row0+lm)|^2, both halves

    // per-accumulator row norms: acc VGPR r holds row m = r + 8*hi
    float sqM[8];
    #pragma unroll
    for (int r = 0; r < 8; ++r) sqM[r] = __shfl(sqLane, r + 8*hi);

    for (int ct = 0; ct < 16; ++ct) {
        const int c0  = wave * 256 + ct * 16;
        const int col = c0 + lm;
        const float cx = pcb[col*3+0], cy = pcb[col*3+1], cz = pcb[col*3+2];
        v2f bm;                           // B 4x16, mirrored K split across lane halves
        bm.x = hi ? cz : cx;
        bm.y = hi ? 0.0f : cy;
        const float sqc = cx*cx + cy*cy + cz*cz;

        v8f acc = {0.f,0.f,0.f,0.f,0.f,0.f,0.f,0.f};
        // D = A x B : 16x16 tile of inner products, K=4 (D=3 + zero pad)
        acc = __builtin_amdgcn_wmma_f32_16x16x4_f32(
            /*neg_a=*/false, a, /*neg_b=*/false, bm,
            /*c_mod=*/(short)0, acc, /*reuse_a=*/false, /*reuse_b=*/false);

        // adj[m][col] = |r_m|^2 + |c|^2 - 2 * <r_m, c>
        #pragma unroll
        for (int r = 0; r < 8; ++r) {
            const int m = r + 8*hi;
            adjS[m * N_ + col] = sqM[r] + sqc - 2.0f * acc[r];
        }
    }
    __syncthreads();

    // Top-20 smallest per row; wave w owns rows 4w..4w+3.
    volatile float* vadj = adjS;
    for (int rr = 0; rr < 4; ++rr) {
        const int row = wave * 4 + rr;
        volatile float* arow = vadj + row * N_;
        int* outIdx = knnIdx + ((size_t)b * N_ + (row0 + row)) * K_;
        for (int t = 0; t < K_; ++t) {
            float bv = FLT_MAX; int bi = 0x7fffffff;
            #pragma unroll
            for (int j = 0; j < 32; ++j) {
                const int ci = lane + 32*j;     // ascending -> lowest idx wins ties
                const float v = arow[ci];
                if (v < bv) { bv = v; bi = ci; }
            }
            #pragma unroll
            for (int off = 16; off > 0; off >>= 1) {
                const float ov = __shfl_xor(bv, off);
                const int   oi = __shfl_xor(bi, off);
                if (ov < bv || (ov == bv && oi < bi)) { bv = ov; bi = oi; }
            }
            if (lane == 0) { outIdx[t] = bi; arow[bi] = FLT_MAX; }
            // same-wave LDS ops are in-order; volatile prevents register caching
        }
    }
}

// ---------------------------------------------------------------------------
// Kernel 2: fused gather + both MLP branches + BN + softmax + outputs.
// Thread = one (point, head). Block = 256 threads = 64 points x 4 heads.
// Neighbor coords staged in LDS; per-head weights folded into registers.
// ---------------------------------------------------------------------------
__global__ __launch_bounds__(256) void attn_fuse_kernel(
    const float* __restrict__ pc,
    const float* __restrict__ Ws1, const float* __restrict__ bs1,
    const float* __restrict__ gs1, const float* __restrict__ betas1,
    const float* __restrict__ Ws2, const float* __restrict__ bs2,
    const float* __restrict__ gs2, const float* __restrict__ betas2,
    const float* __restrict__ Wn1, const float* __restrict__ bn1,
    const float* __restrict__ gn1, const float* __restrict__ betan1,
    const float* __restrict__ Wn2, const float* __restrict__ bn2,
    const float* __restrict__ gn2, const float* __restrict__ betan2,
    const float* __restrict__ out_bias,
    const int* __restrict__ knnIdx,
    float* __restrict__ maf, float* __restrict__ mgf, float* __restrict__ mac)
{
    __shared__ float nbrS[64 * K_ * 3];   // 15 KB
    __shared__ float pcS[64 * 3];

    const int t   = threadIdx.x;
    const int pn0 = blockIdx.x * 64;      // global point index base (flat over B*N)

    for (int e = t; e < 64*3; e += 256) pcS[e] = pc[(size_t)pn0*3 + e];
    for (int e = t; e < 64*K_; e += 256) {
        const int p = e / K_, k = e % K_;
        const int gp = pn0 + p;
        const int bb = gp >> 10;                 // batch (64 pts never straddle)
        const int j  = knnIdx[(size_t)gp * K_ + k];
        const size_t src = ((size_t)bb * N_ + j) * 3;
        nbrS[(p*K_ + k)*3 + 0] = pc[src + 0];
        nbrS[(p*K_ + k)*3 + 1] = pc[src + 1];
        nbrS[(p*K_ + k)*3 + 2] = pc[src + 2];
    }
    __syncthreads();

    const int h  = t & 3;
    const int p  = t >> 2;
    const int gp = pn0 + p;
    const int n  = gp & (N_ - 1);
    const float rs = rsqrtf(1.0f + BN_EPS);

    // fold neighbor-branch head params into registers
    float w0[F_], w1[F_], w2[F_], nb_[F_], nsc[F_], nbt[F_], v2w[F_];
    #pragma unroll
    for (int f = 0; f < F_; ++f) {
        w0[f]  = Wn1[(h*3+0)*F_+f];
        w1[f]  = Wn1[(h*3+1)*F_+f];
        w2[f]  = Wn1[(h*3+2)*F_+f];
        nb_[f] = bn1[h*F_+f];
        nsc[f] = gn1[h*F_+f]*rs;
        nbt[f] = betan1[h*F_+f];
        v2w[f] = Wn2[h*F_+f];
    }
    const float n2b = bn2[h], n2s = gn2[h]*rs, n2t = betan2[h];

    const float px = pcS[p*3+0], py = pcS[p*3+1], pz = pcS[p*3+2];

    // self-attention branch -> scalar sa2
    float accS = 0.0f;
    #pragma unroll
    for (int f = 0; f < F_; ++f) {
        float s = px*Ws1[(h*3+0)*F_+f] + py*Ws1[(h*3+1)*F_+f]
                + pz*Ws1[(h*3+2)*F_+f] + bs1[h*F_+f];
        s = fmaxf(s, 0.0f);
        s = s * (gs1[h*F_+f]*rs) + betas1[h*F_+f];
        accS += s * Ws2[h*F_+f];
    }
    const float sa2 = fmaxf(accS + bs2[h], 0.0f) * (gs2[h]*rs) + betas2[h];

    // pass 1: logits over k
    float lg[K_];
    float mx = -FLT_MAX;
    #pragma unroll
    for (int k = 0; k < K_; ++k) {
        const float dx = px - nbrS[(p*K_+k)*3+0];
        const float dy = py - nbrS[(p*K_+k)*3+1];
        const float dz = pz - nbrS[(p*K_+k)*3+2];
        float a2 = 0.0f;
        #pragma unroll
        for (int f = 0; f < F_; ++f) {
            float v = dx*w0[f] + dy*w1[f] + dz*w2[f] + nb_[f];
            v = fmaxf(v, 0.0f);
            v = v*nsc[f] + nbt[f];
            a2 += v * v2w[f];
        }
        const float na2 = fmaxf(a2 + n2b, 0.0f) * n2s + n2t;
        float L = sa2 + na2;
        L = (L > 0.0f) ? L : 0.2f*L;          // leaky relu
        lg[k] = L;
        mx = fmaxf(mx, L);
    }
    float ssum = 0.0f;
    #pragma unroll
    for (int k = 0; k < K_; ++k) { lg[k] = __expf(lg[k] - mx); ssum += lg[k]; }
    const float inv = 1.0f / ssum;
    #pragma unroll
    for (int k = 0; k < K_; ++k) lg[k] *= inv;    // attn weights

    // mac: [B,N,1,H*K] = attn, head-major
    float4* macv = (float4*)(mac + (size_t)gp * (H_*K_) + h*K_);
    #pragma unroll
    for (int q = 0; q < K_/4; ++q)
        macv[q] = make_float4(lg[4*q+0], lg[4*q+1], lg[4*q+2], lg[4*q+3]);

    // pass 2: recompute na, stream mgf, accumulate attention-weighted features
    float af[F_];
    #pragma unroll
    for (int f = 0; f < F_; ++f) af[f] = 0.0f;
    #pragma unroll
    for (int k = 0; k < K_; ++k) {
        const float dx = px - nbrS[(p*K_+k)*3+0];
        const float dy = py - nbrS[(p*K_+k)*3+1];
        const float dz = pz - nbrS[(p*K_+k)*3+2];
        const float ak = lg[k];
        float vv[F_];
        #pragma unroll
        for (int f = 0; f < F_; ++f) {
            float v = dx*w0[f] + dy*w1[f] + dz*w2[f] + nb_[f];
            v = fmaxf(v, 0.0f);
            v = v*nsc[f] + nbt[f];
            vv[f] = v;
            af[f] += ak * v;
        }
        float4* mgv = (float4*)(mgf + ((size_t)gp * K_ + k) * (H_*F_) + h*F_);
        #pragma unroll
        for (int q = 0; q < 4; ++q)
            mgv[q] = make_float4(vv[4*q+0], vv[4*q+1], vv[4*q+2], vv[4*q+3]);
    }

    // maf: relu(af + out_bias[h,n,0,:]), head-major concat
    const float* ob = out_bias + ((size_t)h * N_ + n) * F_;
    float4* mafv = (float4*)(maf + (size_t)gp * (H_*F_) + h*F_);
    #pragma unroll
    for (int q = 0; q < 4; ++q) {
        mafv[q] = make_float4(
            fmaxf(af[4*q+0] + ob[4*q+0], 0.0f),
            fmaxf(af[4*q+1] + ob[4*q+1], 0.0f),
            fmaxf(af[4*q+2] + ob[4*q+2], 0.0f),
            fmaxf(af[4*q+3] + ob[4*q+3], 0.0f));
    }
}

// ---------------------------------------------------------------------------
extern "C" void kernel_launch(void* const* d_in, const int* in_sizes, int n_in,
                              void* d_out, int out_size, void* d_ws, size_t ws_size,
                              hipStream_t stream)
{
    (void)in_sizes; (void)n_in; (void)out_size; (void)ws_size;

    const float* pc      = (const float*)d_in[0];
    const float* Ws1     = (const float*)d_in[1];
    const float* bs1     = (const float*)d_in[2];
    const float* gs1     = (const float*)d_in[3];
    const float* betas1  = (const float*)d_in[4];
    const float* Ws2     = (const float*)d_in[5];
    const float* bs2     = (const float*)d_in[6];
    const float* gs2     = (const float*)d_in[7];
    const float* betas2  = (const float*)d_in[8];
    const float* Wn1     = (const float*)d_in[9];
    const float* bn1     = (const float*)d_in[10];
    const float* gn1     = (const float*)d_in[11];
    const float* betan1  = (const float*)d_in[12];
    const float* Wn2     = (const float*)d_in[13];
    const float* bn2     = (const float*)d_in[14];
    const float* gn2     = (const float*)d_in[15];
    const float* betan2  = (const float*)d_in[16];
    const float* out_bias= (const float*)d_in[17];

    int* knnIdx = (int*)d_ws;             // B*N*K ints = 5.24 MB scratch

    float* out = (float*)d_out;
    float* maf = out;                                             // B*N*H*F
    float* mgf = maf + (size_t)B_*N_*H_*F_;                       // B*N*K*H*F
    float* mac = mgf + (size_t)B_*N_*K_*H_*F_;                    // B*N*H*K

    dim3 g1(N_/16, B_);
    knn_topk_kernel<<<g1, 128, 0, stream>>>(pc, knnIdx);

    dim3 g2((B_*N_*H_)/256);
    attn_fuse_kernel<<<g2, 256, 0, stream>>>(
        pc, Ws1, bs1, gs1, betas1, Ws2, bs2, gs2, betas2,
        Wn1, bn1, gn1, betan1, Wn2, bn2, gn2, betan2,
        out_bias, knnIdx, maf, mgf, mac);
}